// MAGNN_DGCN_Fusion_16698832847062
// MI455X (gfx1250) — compile-verified
//
#include <hip/hip_runtime.h>
#include <hip/hip_bf16.h>
#include <math.h>

#define N_NODES 100000
#define N_EDGES 1600000
#define D       128
#define ALPHA   0.2f
#define EPS_F   1e-8f

typedef float v2f __attribute__((ext_vector_type(2)));
typedef float v8f __attribute__((ext_vector_type(8)));

// ---- raw LDS byte offset of a __shared__ object (ptrtoint of AS(3) ptr) ----
__device__ __forceinline__ unsigned lds_addr_of(const void* p) {
    return (unsigned)(unsigned long long)(const __attribute__((address_space(3))) char*)p;
}

// ---- order-preserving float <-> uint encoding for atomic segment-max ----
__device__ __forceinline__ unsigned enc_f32(float x) {
    unsigned u = __float_as_uint(x);
    return (u & 0x80000000u) ? ~u : (u | 0x80000000u);
}
__device__ __forceinline__ float dec_f32(unsigned e) {
    unsigned b = (e & 0x80000000u) ? (e & 0x7FFFFFFFu) : ~e;
    return __uint_as_float(b);
}

// ---------------------------------------------------------------------------
// 0) init: zero h_prime accumulator (d_out), reset row_max (encoded 0 == -inf
//    sentinel below all real encodings) and row_sum.
// ---------------------------------------------------------------------------
__global__ void gat_init_kernel(float* __restrict__ out,
                                unsigned* __restrict__ row_max_u,
                                float* __restrict__ row_sum) {
    int i = blockIdx.x * blockDim.x + threadIdx.x;
    if (i < N_NODES) { row_max_u[i] = 0u; row_sum[i] = 0.0f; }
    if (i < N_NODES * D) out[i] = 0.0f;
}

// ---------------------------------------------------------------------------
// 1) h = x @ W via V_WMMA_F32_16X16X4_F32.
//    Block: 256 threads = 8 waves. Block computes a 16x128 tile of h.
//    A-tile staged global->LDS with GLOBAL_LOAD_ASYNC_TO_LDS_B128 (ASYNCcnt).
//    Each wave computes one 16x16 C tile (its own 16-col slice), K-loop step 4.
// ---------------------------------------------------------------------------
__global__ void __launch_bounds__(256)
gat_gemm_kernel(const float* __restrict__ x, const float* __restrict__ W,
                float* __restrict__ h) {
    __shared__ float As[16 * D];                 // 16 rows x 128 K = 8 KB
    const int tid = threadIdx.x;
    const int rowBase = blockIdx.x * 16;         // N_NODES % 16 == 0

    // async stage of the A tile: 8 KB = 512 x b128; 2 x b128 per thread,
    // copied straight into LDS by the async path (no VGPR bounce).
    {
        const char* s0 = (const char*)(x + (size_t)rowBase * D) + tid * 16u;
        unsigned    d0 = lds_addr_of(&As[0]) + tid * 16u;
        asm volatile("global_load_async_to_lds_b128 %0, %1, off"
                     :: "v"(d0), "v"(s0) : "memory");
        asm volatile("global_load_async_to_lds_b128 %0, %1, off"
                     :: "v"(d0 + 4096u), "v"(s0 + 4096u) : "memory");
        asm volatile("s_wait_asynccnt 0x0" ::: "memory");
    }
    __syncthreads();

    const int wave = tid >> 5;                   // 0..7 -> output column tile
    const int lane = tid & 31;
    const int half = lane >> 4;                  // 0: lanes 0-15, 1: lanes 16-31
    const int m    = lane & 15;                  // row (A) / col (B,C) index
    const int n0   = wave * 16;

    v8f c = {};
    #pragma unroll 4
    for (int k = 0; k < D; k += 4) {
        const int kk = k + 2 * half;
        // A 16x4 f32 layout: lane m holds row m; VGPR0/1 = K = kk, kk+1
        v2f a;
        a.x = As[m * D + kk];
        a.y = As[m * D + kk + 1];
        // B 4x16 f32 layout: lane m holds col n0+m; VGPR0/1 = rows K = kk, kk+1
        v2f b;
        b.x = W[(kk)     * D + n0 + m];
        b.y = W[(kk + 1) * D + n0 + m];
        // (neg_a, A, neg_b, B, c_mod, C, reuse_a, reuse_b)
        c = __builtin_amdgcn_wmma_f32_16x16x4_f32(false, a, false, b,
                                                  (short)0, c, false, false);
    }

    // C/D 16x16 f32 layout: VGPR r -> row r (lanes 0-15) / row r+8 (lanes 16-31)
    #pragma unroll
    for (int r = 0; r < 8; ++r) {
        h[(size_t)(rowBase + r + 8 * half) * D + n0 + m] = c[r];
    }
}

// ---------------------------------------------------------------------------
// 2) s_src[i] = h[i,:] . a[:128],  s_dst[i] = h[i,:] . a[128:]
//    One wave per node; lane handles a float4 slice; shuffle reduction.
// ---------------------------------------------------------------------------
__global__ void __launch_bounds__(256)
gat_score_kernel(const float* __restrict__ h, const float* __restrict__ a,
                 float* __restrict__ s_src, float* __restrict__ s_dst) {
    const int wave = threadIdx.x >> 5;
    const int lane = threadIdx.x & 31;
    const int node = blockIdx.x * 8 + wave;      // N_NODES % 8 == 0
    const float4 hv = *(const float4*)(h + (size_t)node * D + lane * 4);
    const float4 as = *(const float4*)(a + lane * 4);
    const float4 ad = *(const float4*)(a + D + lane * 4);
    float ps = hv.x * as.x + hv.y * as.y + hv.z * as.z + hv.w * as.w;
    float pd = hv.x * ad.x + hv.y * ad.y + hv.z * ad.z + hv.w * ad.w;
    #pragma unroll
    for (int off = 16; off > 0; off >>= 1) {
        ps += __shfl_down(ps, off, 32);
        pd += __shfl_down(pd, off, 32);
    }
    if (lane == 0) { s_src[node] = ps; s_dst[node] = pd; }
}

// ---------------------------------------------------------------------------
// 3) e = leaky_relu(s_src[row] + s_dst[col]); segment max via u32 atomicMax
// ---------------------------------------------------------------------------
__global__ void __launch_bounds__(256)
gat_edge_max_kernel(const int* __restrict__ row, const int* __restrict__ col,
                    const float* __restrict__ s_src, const float* __restrict__ s_dst,
                    float* __restrict__ e_buf, unsigned* __restrict__ row_max_u) {
    const int t = blockIdx.x * blockDim.x + threadIdx.x;   // E % 256 == 0
    const int r = row[t];
    const int c = col[t];
    float e = s_src[r] + s_dst[c];
    e = (e > 0.0f) ? e : ALPHA * e;
    e_buf[t] = e;
    atomicMax(&row_max_u[r], enc_f32(e));
}

// ---------------------------------------------------------------------------
// 4) e_exp = exp(e - row_max[row]); segment sum via f32 atomicAdd
// ---------------------------------------------------------------------------
__global__ void __launch_bounds__(256)
gat_edge_exp_kernel(const int* __restrict__ row, float* __restrict__ e_buf,
                    const unsigned* __restrict__ row_max_u,
                    float* __restrict__ row_sum) {
    const int t = blockIdx.x * blockDim.x + threadIdx.x;
    const int r = row[t];
    const float ee = expf(e_buf[t] - dec_f32(row_max_u[r]));
    e_buf[t] = ee;
    atomicAdd(&row_sum[r], ee);
}

// ---------------------------------------------------------------------------
// 5) h_prime[row] += att * h[col]; wave per edge, lane per float4 slice.
//    Gather of h[col] and the scatter atomics both stay in the 192 MB L2.
// ---------------------------------------------------------------------------
__global__ void __launch_bounds__(256)
gat_scatter_kernel(const int* __restrict__ row, const int* __restrict__ col,
                   const float* __restrict__ e_buf, const float* __restrict__ row_sum,
                   const float* __restrict__ h, float* __restrict__ out) {
    const int wave = threadIdx.x >> 5;
    const int lane = threadIdx.x & 31;
    const int eidx = blockIdx.x * 8 + wave;      // E % 8 == 0
    const int r = row[eidx];
    const int c = col[eidx];
    const float att = e_buf[eidx] / fmaxf(row_sum[r], EPS_F);
    const float4 hv = *(const float4*)(h + (size_t)c * D + lane * 4);
    float* dst = out + (size_t)r * D + lane * 4;
    atomicAdd(dst + 0, att * hv.x);
    atomicAdd(dst + 1, att * hv.y);
    atomicAdd(dst + 2, att * hv.z);
    atomicAdd(dst + 3, att * hv.w);
}

// ---------------------------------------------------------------------------
// 6) in-place ELU on the accumulated h_prime
// ---------------------------------------------------------------------------
__global__ void gat_elu_kernel(float* __restrict__ out) {
    const int i = blockIdx.x * blockDim.x + threadIdx.x;
    if (i < N_NODES * D) {
        const float v = out[i];
        out[i] = (v > 0.0f) ? v : expm1f(v);
    }
}

extern "C" void kernel_launch(void* const* d_in, const int* in_sizes, int n_in,
                              void* d_out, int out_size, void* d_ws, size_t ws_size,
                              hipStream_t stream) {
    const float* x   = (const float*)d_in[0];   // [N,128]
    const int*   row = (const int*)d_in[1];     // [E]
    const int*   col = (const int*)d_in[2];     // [E]
    const float* W   = (const float*)d_in[3];   // [128,128]
    const float* a   = (const float*)d_in[4];   // [256,1]
    float*       out = (float*)d_out;           // [N,128] (h_prime accumulator)

    char* ws = (char*)d_ws;
    float*    h         = (float*)ws;    ws += (size_t)N_NODES * D * sizeof(float);
    float*    s_src     = (float*)ws;    ws += (size_t)N_NODES * sizeof(float);
    float*    s_dst     = (float*)ws;    ws += (size_t)N_NODES * sizeof(float);
    unsigned* row_max_u = (unsigned*)ws; ws += (size_t)N_NODES * sizeof(unsigned);
    float*    row_sum   = (float*)ws;    ws += (size_t)N_NODES * sizeof(float);
    float*    e_buf     = (float*)ws;    ws += (size_t)N_EDGES * sizeof(float);

    gat_init_kernel   <<<(N_NODES * D + 255) / 256, 256, 0, stream>>>(out, row_max_u, row_sum);
    gat_gemm_kernel   <<<N_NODES / 16,              256, 0, stream>>>(x, W, h);
    gat_score_kernel  <<<N_NODES / 8,               256, 0, stream>>>(h, a, s_src, s_dst);
    gat_edge_max_kernel<<<N_EDGES / 256,            256, 0, stream>>>(row, col, s_src, s_dst, e_buf, row_max_u);
    gat_edge_exp_kernel<<<N_EDGES / 256,            256, 0, stream>>>(row, e_buf, row_max_u, row_sum);
    gat_scatter_kernel<<<N_EDGES / 8,               256, 0, stream>>>(row, col, e_buf, row_sum, h, out);
    gat_elu_kernel    <<<(N_NODES * D + 255) / 256, 256, 0, stream>>>(out);
}